// MetaLearningCrisisMemory_10720238371059
// MI455X (gfx1250) — compile-verified
//
#include <hip/hip_runtime.h>
#include <hip/hip_bf16.h>
#include <math.h>

// ---------------------------------------------------------------------------
// MetaLearningCrisisMemory — MI455X (gfx1250) implementation.
// Bandwidth-bound after algebraic collapse of the two memory-bank GEMMs:
//   scores = (per-head folded query s[8,1024]) @ memory_keys^T   (205 MB stream)
//   u[h]   = sum_m attn[h,m] * memory_values[m]                  (410 MB stream)
// Both use V_WMMA_F32_16X16X4_F32 (fp32 WMMA, heads padded 8->16).
// Stationary WMMA operands are staged into LDS with GLOBAL_LOAD_ASYNC_TO_LDS.
// ---------------------------------------------------------------------------

#define IN_DIM   4096
#define HID      2048
#define H2       1024
#define MEMN     50000
#define NH       8
#define DQ       128
#define PN       64
#define TOPK     5
#define EPSV     1e-5f
#define INV_SQRT_DQ 0.08838834764831845f  // 1/sqrt(128)

typedef __attribute__((ext_vector_type(2))) float v2f;
typedef __attribute__((ext_vector_type(8))) float v8f;

#if defined(__AMDGCN__) && \
    __has_builtin(__builtin_amdgcn_wmma_f32_16x16x4_f32)
#define HAVE_WMMA_F32 1
#endif

#if defined(__AMDGCN__) && \
    __has_builtin(__builtin_amdgcn_global_load_async_to_lds_b128) && \
    __has_builtin(__builtin_amdgcn_s_wait_asynccnt)
#define HAVE_ASYNC_LDS 1
#endif

#ifdef HAVE_ASYNC_LDS
// Builtin signature (from hipcc diagnostic): first param is
// 'int __vector(4) __device__ *' i.e. addrspace(1) v4i*, second the
// addrspace(3) counterpart. as(3) pointers are 32-bit LDS byte offsets ==
// low 32 bits of the flat address (ISA: LDS aperture maps addr[31:0]).
typedef int v4i_vs __attribute__((vector_size(16)));
__device__ __forceinline__ __attribute__((address_space(1))) v4i_vs* glob_v4(const void* p) {
  return (__attribute__((address_space(1))) v4i_vs*)(unsigned long long)p;
}
__device__ __forceinline__ __attribute__((address_space(3))) v4i_vs* lds_v4(void* p) {
  return (__attribute__((address_space(3))) v4i_vs*)(unsigned int)(unsigned long long)p;
}
#endif

// ---- workspace layout (float offsets) -------------------------------------
#define OFF_COMBINED 0          /* 6144: [enc | attended | proto_pad] */
#define OFF_HB       6144       /* 2048 */
#define OFF_T1       8192       /* 2048 (reused as ctx) */
#define OFF_Q        10240      /* 1024 */
#define OFF_QH       11264      /* 1024 */
#define OFF_S16      12288      /* 16*1024 folded query (scaled) */
#define OFF_CH       28672      /* 16  per-head bias const (scaled) */
#define OFF_U        28688      /* 8*2048 attention-weighted value sum */
#define OFF_MAXP     45072      /* 64*8 */
#define OFF_GMAX     45584      /* 8 */
#define OFF_SUMP     45592      /* 64*8 */
#define OFF_Z        46104      /* 8 */
#define OFF_BTOP     46112      /* 50*5 (pad 256) */
#define OFF_TOP5     46368      /* 8 */
#define OFF_DIST     46376      /* 64 */
#define OFF_PCONF    46440      /* 8 */
#define OFF_M1       46448      /* 2048 */
#define OFF_M2       48496      /* 1024 */
#define OFF_META     49520      /* 16 */
#define OFF_SCORES   49536      /* 50000*8 */
#define OFF_W16      449536     /* 50000*16 (heads padded to 16, zeros) */
#define OFF_ATTNW    1249536    /* 50000 */
// total ~1.30M floats (~5.2 MB)

// ---------------------------------------------------------------------------
__device__ __forceinline__ float wave_reduce_add(float v) {
  for (int off = 16; off > 0; off >>= 1) v += __shfl_down(v, off, 32);
  return v;
}

__global__ void zero_kernel(float* p, int n) {
  int i = blockIdx.x * blockDim.x + threadIdx.x;
  if (i < n) p[i] = 0.f;
}

// Generic row-parallel matvec: y = f(W x + b); mode 0: none, 1: relu,
// 2: relu then BatchNorm(eval).
__global__ void matvec_kernel(const float* __restrict__ W, const float* __restrict__ x,
                              const float* __restrict__ b, float* __restrict__ y,
                              int rows, int cols, int mode,
                              const float* __restrict__ g, const float* __restrict__ bb,
                              const float* __restrict__ mu, const float* __restrict__ var) {
  int row = blockIdx.x * (blockDim.x >> 5) + (threadIdx.x >> 5);
  int lane = threadIdx.x & 31;
  if (row >= rows) return;
  const float* wr = W + (size_t)row * cols;
  float acc = 0.f;
  for (int c = lane * 4; c < cols; c += 128) {
    float4 wv = *(const float4*)(wr + c);
    float4 xv = *(const float4*)(x + c);
    acc += wv.x * xv.x + wv.y * xv.y + wv.z * xv.z + wv.w * xv.w;
  }
  acc = wave_reduce_add(acc);
  if (lane == 0) {
    float r = acc + b[row];
    if (mode >= 1) r = fmaxf(r, 0.f);
    if (mode == 2) r = (r - mu[row]) * rsqrtf(var[row] + EPSV) * g[row] + bb[row];
    y[row] = r;
  }
}

// s16[h][j] = (1/sqrt(dq)) * sum_d qh[h*128+d] * wk[h*128+d][j]  (h<8; zero pad h>=8)
__global__ void build_s_kernel(const float* __restrict__ qh, const float* __restrict__ wk,
                               float* __restrict__ s16) {
  int idx = blockIdx.x * blockDim.x + threadIdx.x;   // 16*1024 threads
  int h = idx >> 10, j = idx & 1023;
  float acc = 0.f;
  if (h < NH) {
    for (int d = 0; d < DQ; ++d) acc += qh[h * DQ + d] * wk[(size_t)(h * DQ + d) * H2 + j];
    acc *= INV_SQRT_DQ;
  }
  s16[idx] = acc;
}

// chead[h] = (1/sqrt(dq)) * qh_h . bk_h  (h<8)
__global__ void build_chead_kernel(const float* __restrict__ qh, const float* __restrict__ bk,
                                   float* __restrict__ chead) {
  int h = threadIdx.x >> 5, lane = threadIdx.x & 31;
  float acc = 0.f;
  for (int d = lane; d < DQ; d += 32) acc += qh[h * DQ + d] * bk[h * DQ + d];
  acc = wave_reduce_add(acc);
  if (lane == 0) chead[h] = acc * INV_SQRT_DQ;
  if (threadIdx.x >= 8 && threadIdx.x < 16) chead[threadIdx.x + 8] = 0.f; // chead[16..23]=0
}

// scores[m][h] = s16[h] . mk[m] + chead[h]   via V_WMMA_F32_16X16X4_F32
// D(16 mem rows x 16 heads) = A(mk 16x4) * B(s^T 4x16), K-loop over 1024.
// B rows (8 real heads + shared zero row) staged in LDS via async copy.
__global__ __launch_bounds__(128) void scores_wmma_kernel(const float* __restrict__ mk,
                                                          const float* __restrict__ s16,
                                                          const float* __restrict__ chead,
                                                          float* __restrict__ scores) {
  __shared__ float sB[9 * H2];                 // 8 head rows + 1 zero row (36 KB)
  int wave = threadIdx.x >> 5, lane = threadIdx.x & 31;
  int ln15 = lane & 15, kg = lane >> 4;

#ifdef HAVE_ASYNC_LDS
  for (int i = threadIdx.x * 4; i < 8 * H2; i += 128 * 4)
    __builtin_amdgcn_global_load_async_to_lds_b128(glob_v4(s16 + i), lds_v4(&sB[i]), 0, 0);
  for (int i = threadIdx.x; i < H2; i += 128) sB[8 * H2 + i] = 0.f;
  __builtin_amdgcn_s_wait_asynccnt(0);
#else
  for (int i = threadIdx.x * 4; i < 8 * H2; i += 128 * 4)
    *(float4*)(sB + i) = *(const float4*)(s16 + i);
  for (int i = threadIdx.x; i < H2; i += 128) sB[8 * H2 + i] = 0.f;
#endif
  __syncthreads();

  long tile = (long)blockIdx.x * 4 + wave;
  long m0 = tile * 16;
  if (m0 >= MEMN) return;                       // wave-uniform: EXEC stays all-ones
  v8f c = {0.f, 0.f, 0.f, 0.f, 0.f, 0.f, 0.f, 0.f};
#ifdef HAVE_WMMA_F32
  // A layout (32-bit 16x4): lane<16 -> row M=lane, K={0,1}; lane+16 -> K={2,3}
  const float* arow = mk + (m0 + ln15) * H2 + 2 * kg;
  const float* brow = sB + (ln15 < NH ? ln15 : NH) * H2 + 2 * kg;  // padded heads -> zero row
  for (int k0 = 0; k0 < H2; k0 += 4) {
    if ((k0 & 63) == 0) __builtin_prefetch(arow + k0 + 256, 0, 1);
    v2f a = *(const v2f*)(arow + k0);
    v2f b = *(const v2f*)(brow + k0);
    c = __builtin_amdgcn_wmma_f32_16x16x4_f32(false, a, false, b, (short)0, c,
                                              false, false);
  }
#else
  for (int r = 0; r < 8; ++r) {
    long m = m0 + r + 8 * kg;
    if (ln15 < NH) {
      const float* mr = mk + m * H2;
      const float* sr = sB + ln15 * H2;
      float acc = 0.f;
      for (int k = 0; k < H2; ++k) acc += mr[k] * sr[k];
      c[r] = acc;
    }
  }
#endif
  float ch = chead[ln15];
  for (int r = 0; r < 8; ++r) {
    long m = m0 + r + 8 * kg;                     // D: VGPR r -> M = r + 8*(lane/16)
    if (ln15 < NH) scores[m * NH + ln15] = c[r] + ch;
  }
}

// per-head max reduction over M (partial per block)
__global__ void maxred_kernel(const float* __restrict__ scores, float* __restrict__ part) {
  __shared__ float sm[256 * NH];
  int tid = threadIdx.x;
  float mx[NH];
  for (int h = 0; h < NH; ++h) mx[h] = -3.4e38f;
  for (int m = blockIdx.x * 256 + tid; m < MEMN; m += 64 * 256)
    for (int h = 0; h < NH; ++h) mx[h] = fmaxf(mx[h], scores[(long)m * NH + h]);
  for (int h = 0; h < NH; ++h) sm[h * 256 + tid] = mx[h];
  __syncthreads();
  for (int s = 128; s > 0; s >>= 1) {
    if (tid < s)
      for (int h = 0; h < NH; ++h)
        sm[h * 256 + tid] = fmaxf(sm[h * 256 + tid], sm[h * 256 + tid + s]);
    __syncthreads();
  }
  if (tid < NH) part[blockIdx.x * NH + tid] = sm[tid * 256];
}

__global__ void sumred_kernel(const float* __restrict__ scores, const float* __restrict__ gmax,
                              float* __restrict__ part) {
  __shared__ float sm[256 * NH];
  int tid = threadIdx.x;
  float gm[NH], s[NH];
  for (int h = 0; h < NH; ++h) { gm[h] = gmax[h]; s[h] = 0.f; }
  for (int m = blockIdx.x * 256 + tid; m < MEMN; m += 64 * 256)
    for (int h = 0; h < NH; ++h) s[h] += expf(scores[(long)m * NH + h] - gm[h]);
  for (int h = 0; h < NH; ++h) sm[h * 256 + tid] = s[h];
  __syncthreads();
  for (int st = 128; st > 0; st >>= 1) {
    if (tid < st)
      for (int h = 0; h < NH; ++h) sm[h * 256 + tid] += sm[h * 256 + tid + st];
    __syncthreads();
  }
  if (tid < NH) part[blockIdx.x * NH + tid] = sm[tid * 256];
}

__global__ void fin8_kernel(const float* __restrict__ part, float* __restrict__ outp,
                            int nblk, int isSum) {
  int h = threadIdx.x;
  if (h >= NH) return;
  float v = isSum ? 0.f : -3.4e38f;
  for (int b = 0; b < nblk; ++b) {
    float x = part[b * NH + h];
    v = isSum ? (v + x) : fmaxf(v, x);
  }
  outp[h] = v;
}

// attn weights (padded to 16 heads, zeros) + head-averaged weight
__global__ void weights_kernel(const float* __restrict__ scores, const float* __restrict__ gmax,
                               const float* __restrict__ Z, float* __restrict__ w16,
                               float* __restrict__ attnw) {
  int m = blockIdx.x * blockDim.x + threadIdx.x;
  if (m >= MEMN) return;
  float s = 0.f;
  for (int h = 0; h < NH; ++h) {
    float w = expf(scores[(long)m * NH + h] - gmax[h]) / Z[h];
    w16[(long)m * 16 + h] = w;
    s += w;
  }
  for (int h = NH; h < 16; ++h) w16[(long)m * 16 + h] = 0.f;
  attnw[m] = s * (1.0f / NH);
}

// u[h][n] += sum_m w16[m][h] * mv[m][n]   via V_WMMA_F32_16X16X4_F32
// D(16 heads x 16 cols) = A(w^T 16x4) * B(mv 4x16), K = memory index.
// A chunk (400x16 weights, 25.6 KB) staged in LDS via async copy.
__global__ __launch_bounds__(128) void u_wmma_kernel(const float* __restrict__ mv,
                                                     const float* __restrict__ w16,
                                                     float* __restrict__ u) {
  __shared__ float sW[400 * 16];
  int wave = threadIdx.x >> 5, lane = threadIdx.x & 31;
  int ln15 = lane & 15, kg = lane >> 4;
  int n0 = blockIdx.x * 64 + wave * 16;
  long mbeg = (long)blockIdx.y * 400;
  const float* wsrc = w16 + mbeg * 16;

#ifdef HAVE_ASYNC_LDS
  for (int i = threadIdx.x * 4; i < 400 * 16; i += 128 * 4)
    __builtin_amdgcn_global_load_async_to_lds_b128(glob_v4(wsrc + i), lds_v4(&sW[i]), 0, 0);
  __builtin_amdgcn_s_wait_asynccnt(0);
#else
  for (int i = threadIdx.x * 4; i < 400 * 16; i += 128 * 4)
    *(float4*)(sW + i) = *(const float4*)(wsrc + i);
#endif
  __syncthreads();

  v8f c = {0.f, 0.f, 0.f, 0.f, 0.f, 0.f, 0.f, 0.f};
#ifdef HAVE_WMMA_F32
  for (int k = 0; k < 400; k += 4) {
    int m = k + 2 * kg;                       // chunk-local row
    if ((k & 31) == 0) __builtin_prefetch(mv + (mbeg + m + 8) * HID + n0, 0, 1);
    v2f a = {sW[m * 16 + ln15], sW[(m + 1) * 16 + ln15]};
    v2f b = {mv[(mbeg + m) * HID + n0 + ln15], mv[(mbeg + m + 1) * HID + n0 + ln15]};
    c = __builtin_amdgcn_wmma_f32_16x16x4_f32(false, a, false, b, (short)0, c,
                                              false, false);
  }
#else
  if (kg == 0) {
    for (int r = 0; r < 8; ++r) {
      float acc = 0.f;
      for (int k = 0; k < 400; ++k)
        acc += sW[k * 16 + r] * mv[(mbeg + k) * HID + n0 + ln15];
      c[r] = acc;
    }
  }
#endif
  if (kg == 0)   // heads 0..7 live in lanes 0-15 (rows M=r); padded heads are zero
    for (int r = 0; r < 8; ++r)
      atomicAdd(&u[r * HID + n0 + ln15], c[r]);
}

// ctx[i] = wv[i] . u[i/256] + bv[i]  (per-head value projection of weighted sum)
__global__ void ctx_matvec_kernel(const float* __restrict__ wvm, const float* __restrict__ u,
                                  const float* __restrict__ bv, float* __restrict__ ctx) {
  int row = blockIdx.x * 8 + (threadIdx.x >> 5);
  int lane = threadIdx.x & 31;
  if (row >= HID) return;
  const float* x = u + (size_t)(row >> 8) * HID;
  const float* wr = wvm + (size_t)row * HID;
  float acc = 0.f;
  for (int cix = lane * 4; cix < HID; cix += 128) {
    float4 wv4 = *(const float4*)(wr + cix);
    float4 xv4 = *(const float4*)(x + cix);
    acc += wv4.x * xv4.x + wv4.y * xv4.y + wv4.z * xv4.z + wv4.w * xv4.w;
  }
  acc = wave_reduce_add(acc);
  if (lane == 0) ctx[row] = acc + bv[row];
}

__device__ __forceinline__ void top5_insert(float* t, float v) {
  if (v <= t[4]) return;
  int i = 4;
  while (i > 0 && t[i - 1] < v) { t[i] = t[i - 1]; --i; }
  t[i] = v;
}

__global__ void top_part_kernel(const float* __restrict__ attnw, float* __restrict__ btop) {
  __shared__ float buf[256 * 5];
  int tid = threadIdx.x;
  float t[5] = {-3.4e38f, -3.4e38f, -3.4e38f, -3.4e38f, -3.4e38f};
  for (int m = blockIdx.x * 256 + tid; m < MEMN; m += 50 * 256) top5_insert(t, attnw[m]);
  for (int i = 0; i < 5; ++i) buf[tid * 5 + i] = t[i];
  __syncthreads();
  if (tid == 0) {
    float f[5] = {-3.4e38f, -3.4e38f, -3.4e38f, -3.4e38f, -3.4e38f};
    for (int i = 0; i < 256 * 5; ++i) top5_insert(f, buf[i]);
    for (int i = 0; i < 5; ++i) btop[blockIdx.x * 5 + i] = f[i];
  }
}

__global__ void top_fin_kernel(const float* __restrict__ btop, float* __restrict__ top5) {
  if (threadIdx.x != 0) return;
  float f[5] = {-3.4e38f, -3.4e38f, -3.4e38f, -3.4e38f, -3.4e38f};
  for (int i = 0; i < 50 * 5; ++i) top5_insert(f, btop[i]);
  for (int i = 0; i < 5; ++i) top5[i] = f[i];
}

__global__ void proto_dist_kernel(const float* __restrict__ enc, const float* __restrict__ protos,
                                  float* __restrict__ dist) {
  __shared__ float sm[256];
  int p = blockIdx.x, tid = threadIdx.x;
  float acc = 0.f;
  for (int j = tid; j < H2; j += 256) {
    float d = enc[j] - protos[(size_t)p * H2 + j];
    acc += d * d;
  }
  sm[tid] = acc;
  __syncthreads();
  for (int s = 128; s > 0; s >>= 1) {
    if (tid < s) sm[tid] += sm[tid + s];
    __syncthreads();
  }
  if (tid == 0) dist[p] = sqrtf(sm[0]);
}

__global__ void proto_fin_kernel(const float* __restrict__ dist, const float* __restrict__ protos,
                                 float* __restrict__ combined, float* __restrict__ pconf) {
  __shared__ int sidx;
  if (threadIdx.x == 0) {
    int bi = 0;
    float bv = dist[0];
    for (int p = 1; p < PN; ++p)
      if (dist[p] < bv) { bv = dist[p]; bi = p; }
    sidx = bi;
    *pconf = 1.f / (1.f + bv);
  }
  __syncthreads();
  for (int j = threadIdx.x; j < H2; j += 256) {
    combined[2 * HID + j] = protos[(size_t)sidx * H2 + j];
    combined[2 * HID + H2 + j] = 0.f;   // zero pad H2 -> HID
  }
}

__global__ void head_kernel(const float* __restrict__ meta, const float* __restrict__ pconf,
                            const float* __restrict__ top5, float* __restrict__ out) {
  if (threadIdx.x != 0 || blockIdx.x != 0) return;
  float a = meta[0], b = meta[1], c = meta[2];
  float mx = fmaxf(a, fmaxf(b, c));
  float ea = expf(a - mx), eb = expf(b - mx), ec = expf(c - mx);
  float s = ea + eb + ec;
  out[0] = ea / s; out[1] = eb / s; out[2] = ec / s;
  out[3] = 1.f / (1.f + expf(-meta[3]));
  out[4] = *pconf;
  for (int i = 0; i < TOPK; ++i) out[5 + i] = top5[i];
}

__global__ void pack_kernel(const float* __restrict__ combined, float* __restrict__ out) {
  int i = blockIdx.x * blockDim.x + threadIdx.x;
  if (i < 3 * HID) out[10 + i] = combined[i];
}

// ---------------------------------------------------------------------------
extern "C" void kernel_launch(void* const* d_in, const int* in_sizes, int n_in,
                              void* d_out, int out_size, void* d_ws, size_t ws_size,
                              hipStream_t stream) {
  const float* obs   = (const float*)d_in[0];
  const float* ce_w1 = (const float*)d_in[1];
  const float* ce_b1 = (const float*)d_in[2];
  const float* bn1_g = (const float*)d_in[3];
  const float* bn1_b = (const float*)d_in[4];
  const float* bn1_m = (const float*)d_in[5];
  const float* bn1_v = (const float*)d_in[6];
  const float* ce_w2 = (const float*)d_in[7];
  const float* ce_b2 = (const float*)d_in[8];
  const float* qe_w1 = (const float*)d_in[9];
  const float* qe_b1 = (const float*)d_in[10];
  const float* qe_w2 = (const float*)d_in[11];
  const float* qe_b2 = (const float*)d_in[12];
  const float* wq    = (const float*)d_in[13];
  const float* bq    = (const float*)d_in[14];
  const float* wk    = (const float*)d_in[15];
  const float* bk    = (const float*)d_in[16];
  const float* wv    = (const float*)d_in[17];
  const float* bv    = (const float*)d_in[18];
  const float* wo    = (const float*)d_in[19];
  const float* bo    = (const float*)d_in[20];
  const float* mkeys = (const float*)d_in[21];
  const float* mvals = (const float*)d_in[22];
  const float* protos= (const float*)d_in[23];
  const float* mp_w1 = (const float*)d_in[24];
  const float* mp_b1 = (const float*)d_in[25];
  const float* bn2_g = (const float*)d_in[26];
  const float* bn2_b = (const float*)d_in[27];
  const float* bn2_m = (const float*)d_in[28];
  const float* bn2_v = (const float*)d_in[29];
  const float* mp_w2 = (const float*)d_in[30];
  const float* mp_b2 = (const float*)d_in[31];
  const float* mp_w3 = (const float*)d_in[32];
  const float* mp_b3 = (const float*)d_in[33];

  float* ws  = (float*)d_ws;
  float* out = (float*)d_out;

  float* combined = ws + OFF_COMBINED;         // enc | attended | proto_pad
  float* enc = combined;                        // [HID]
  float* attended = combined + HID;             // [HID]
  float* hb  = ws + OFF_HB;
  float* t1  = ws + OFF_T1;                     // also ctx
  float* q   = ws + OFF_Q;
  float* qh  = ws + OFF_QH;
  float* s16 = ws + OFF_S16;
  float* ch  = ws + OFF_CH;
  float* u   = ws + OFF_U;
  float* maxp = ws + OFF_MAXP;
  float* gmax = ws + OFF_GMAX;
  float* sump = ws + OFF_SUMP;
  float* Zs   = ws + OFF_Z;
  float* btop = ws + OFF_BTOP;
  float* top5 = ws + OFF_TOP5;
  float* dist = ws + OFF_DIST;
  float* pconf = ws + OFF_PCONF;
  float* m1   = ws + OFF_M1;
  float* m2   = ws + OFF_M2;
  float* meta = ws + OFF_META;
  float* scores = ws + OFF_SCORES;
  float* w16    = ws + OFF_W16;
  float* attnw  = ws + OFF_ATTNW;

  // u accumulators must be zero each call (atomicAdd partials)
  zero_kernel<<<(NH * HID + 255) / 256, 256, 0, stream>>>(u, NH * HID);

  // crisis encoder: h = BN(relu(W1 x + b1)); enc = W2 h + b2
  matvec_kernel<<<HID / 8, 256, 0, stream>>>(ce_w1, obs, ce_b1, hb, HID, IN_DIM, 2,
                                             bn1_g, bn1_b, bn1_m, bn1_v);
  matvec_kernel<<<HID / 8, 256, 0, stream>>>(ce_w2, hb, ce_b2, enc, HID, HID, 0,
                                             nullptr, nullptr, nullptr, nullptr);
  // query encoder
  matvec_kernel<<<HID / 8, 256, 0, stream>>>(qe_w1, enc, qe_b1, t1, HID, HID, 1,
                                             nullptr, nullptr, nullptr, nullptr);
  matvec_kernel<<<H2 / 8, 256, 0, stream>>>(qe_w2, t1, qe_b2, q, H2, HID, 0,
                                            nullptr, nullptr, nullptr, nullptr);
  matvec_kernel<<<H2 / 8, 256, 0, stream>>>(wq, q, bq, qh, H2, H2, 0,
                                            nullptr, nullptr, nullptr, nullptr);
  // fold query through Wk: s[h] = qh_h^T Wk_h (scaled), c[h] = qh_h.bk_h (scaled)
  build_s_kernel<<<16 * H2 / 256, 256, 0, stream>>>(qh, wk, s16);
  build_chead_kernel<<<1, 256, 0, stream>>>(qh, bk, ch);

  // scores over 50k memories (fp32 WMMA, streams memory_keys once)
  scores_wmma_kernel<<<(MEMN / 16 + 3) / 4, 128, 0, stream>>>(mkeys, s16, ch, scores);

  // softmax statistics
  maxred_kernel<<<64, 256, 0, stream>>>(scores, maxp);
  fin8_kernel<<<1, 8, 0, stream>>>(maxp, gmax, 64, 0);
  sumred_kernel<<<64, 256, 0, stream>>>(scores, gmax, sump);
  fin8_kernel<<<1, 8, 0, stream>>>(sump, Zs, 64, 1);
  weights_kernel<<<(MEMN + 255) / 256, 256, 0, stream>>>(scores, gmax, Zs, w16, attnw);

  // top-5 of head-averaged attention
  top_part_kernel<<<50, 256, 0, stream>>>(attnw, btop);
  top_fin_kernel<<<1, 32, 0, stream>>>(btop, top5);

  // u[h] = sum_m attn[h,m] mv[m]  (fp32 WMMA, streams memory_values once)
  {
    dim3 g(HID / 64, MEMN / 400);
    u_wmma_kernel<<<g, 128, 0, stream>>>(mvals, w16, u);
  }
  // ctx = per-head Wv u + bv ; attended = Wo ctx + bo
  ctx_matvec_kernel<<<HID / 8, 256, 0, stream>>>(wv, u, bv, t1);
  matvec_kernel<<<HID / 8, 256, 0, stream>>>(wo, t1, bo, attended, HID, HID, 0,
                                             nullptr, nullptr, nullptr, nullptr);
  // prototype matching on enc[:H2]
  proto_dist_kernel<<<PN, 256, 0, stream>>>(enc, protos, dist);
  proto_fin_kernel<<<1, 256, 0, stream>>>(dist, protos, combined, pconf);

  // meta predictor
  matvec_kernel<<<HID / 8, 256, 0, stream>>>(mp_w1, combined, mp_b1, m1, HID, 3 * HID, 2,
                                             bn2_g, bn2_b, bn2_m, bn2_v);
  matvec_kernel<<<H2 / 8, 256, 0, stream>>>(mp_w2, m1, mp_b2, m2, H2, HID, 1,
                                            nullptr, nullptr, nullptr, nullptr);
  matvec_kernel<<<1, 256, 0, stream>>>(mp_w3, m2, mp_b3, meta, 4, H2, 0,
                                       nullptr, nullptr, nullptr, nullptr);

  // final packing
  head_kernel<<<1, 32, 0, stream>>>(meta, pconf, top5, out);
  pack_kernel<<<(3 * HID + 255) / 256, 256, 0, stream>>>(combined, out);
}